// PSRNet_17686675324958
// MI455X (gfx1250) — compile-verified
//
#include <hip/hip_runtime.h>

#define D_DIM 128   // node feature dim
#define H_DIM 64    // hidden dim of top_net

typedef float v2f __attribute__((ext_vector_type(2)));
typedef float v8f __attribute__((ext_vector_type(8)));

// ---------------- Kernel 0: zero sum/count accumulators ----------------
__global__ void psr_zero(float* __restrict__ p, int n) {
    int i = blockIdx.x * blockDim.x + threadIdx.x;
    if (i < n) p[i] = 0.0f;
}

// ---------------- Kernel 1: segmented sum + counts (sorted batch) ------
// One wave per 512-row chunk. Lane owns 4 columns (float4). Register
// accumulation; global f32 atomics only at segment boundaries / chunk end.
__global__ __launch_bounds__(256) void psr_segsum(
        const float* __restrict__ x,
        const long long* __restrict__ batch,
        float* __restrict__ sums,     // [G, 128]
        float* __restrict__ counts,   // [G]
        int n, int rows_per_wave) {
    const int lane = threadIdx.x & 31;
    const int wave = blockIdx.x * (blockDim.x >> 5) + (threadIdx.x >> 5);
    long long r0 = (long long)wave * rows_per_wave;
    if (r0 >= n) return;
    long long r1 = r0 + rows_per_wave;
    if (r1 > n) r1 = n;

    const float4* __restrict__ x4 = (const float4*)x;
    float4 acc = make_float4(0.f, 0.f, 0.f, 0.f);
    int cur = (int)batch[r0];
    int run = 0;

    for (long long base = r0; base < r1; base += 32) {
        int m = (int)(r1 - base);
        if (m > 32) m = 32;
        int bv = 0;
        if (lane < m) bv = (int)batch[base + lane];   // one 8B load / 32 rows
        for (int j = 0; j < m; ++j) {
            int b = __shfl(bv, j, 32);                // wave-uniform broadcast
            if (b != cur) {                           // uniform branch
                float* s = sums + (long long)cur * D_DIM + lane * 4;
                atomicAdd(s + 0, acc.x);
                atomicAdd(s + 1, acc.y);
                atomicAdd(s + 2, acc.z);
                atomicAdd(s + 3, acc.w);
                if (lane == 0) atomicAdd(counts + cur, (float)run);
                acc = make_float4(0.f, 0.f, 0.f, 0.f);
                run = 0;
                cur = b;
            }
            float4 v = x4[(base + j) * (D_DIM / 4) + lane];  // coalesced b128
            acc.x += v.x; acc.y += v.y; acc.z += v.z; acc.w += v.w;
            run++;
        }
    }
    float* s = sums + (long long)cur * D_DIM + lane * 4;
    atomicAdd(s + 0, acc.x);
    atomicAdd(s + 1, acc.y);
    atomicAdd(s + 2, acc.z);
    atomicAdd(s + 3, acc.w);
    if (lane == 0) atomicAdd(counts + cur, (float)run);
}

// ---------------- Kernel 2: mean -> relu(mean@W1 + b1) via f32 WMMA ----
// One wave per 16x16 output tile of H[G, 64]. K loop of 32 x WMMA_F32_16X16X4_F32.
// A-frag layout (32-bit A 16x4): lane l: row = l%16; VGPR0 = K(k + 2*(l/16)),
// VGPR1 = K(k + 2*(l/16) + 1). B-frag mirrors with N across lanes.
// D layout: VGPR v, lane l: M = 8*(l/16)+v, N = l%16.
__global__ __launch_bounds__(256) void psr_mlp1_wmma(
        const float* __restrict__ sums,    // [G, 128]
        const float* __restrict__ counts,  // [G]
        const float* __restrict__ W1,      // [128, 64] row-major
        const float* __restrict__ b1,      // [64]
        float* __restrict__ H,             // [G, 64]
        int mTiles) {
    const int wave  = blockIdx.x * (blockDim.x >> 5) + (threadIdx.x >> 5);
    const int tileM = wave >> 2;   // 16 tiles over G=256
    const int tileN = wave & 3;    // 4 tiles over 64
    if (tileM >= mTiles) return;   // wave-uniform; launched exactly, never taken
    const int lane = threadIdx.x & 31;
    const int half = lane >> 4;
    const int lrow = lane & 15;

    const int arow = tileM * 16 + lrow;
    const float inv = 1.0f / counts[arow];
    const float* __restrict__ srow = sums + (long long)arow * D_DIM;
    const int ncol = tileN * 16 + lrow;

    v8f c = {0.f, 0.f, 0.f, 0.f, 0.f, 0.f, 0.f, 0.f};
    for (int k = 0; k < D_DIM; k += 4) {
        const int ka = k + 2 * half;
        v2f a, b;
        a[0] = srow[ka + 0] * inv;
        a[1] = srow[ka + 1] * inv;
        b[0] = W1[(ka + 0) * H_DIM + ncol];
        b[1] = W1[(ka + 1) * H_DIM + ncol];
        c = __builtin_amdgcn_wmma_f32_16x16x4_f32(
                /*neg_a=*/false, a, /*neg_b=*/false, b,
                /*c_mod=*/(short)0, c, /*reuse_a=*/false, /*reuse_b=*/false);
    }

    const float bias = b1[ncol];
    #pragma unroll
    for (int v = 0; v < 8; ++v) {
        float h = c[v] + bias;
        h = h > 0.f ? h : 0.f;                 // ReLU
        const int M = tileM * 16 + half * 8 + v;
        H[M * H_DIM + ncol] = h;
    }
}

// ---------------- Kernel 3: out = H @ W2 + b2 (wave-reduce matvec) -----
__global__ void psr_mlp2(const float* __restrict__ H,
                         const float* __restrict__ W2,   // [64]
                         const float* __restrict__ b2,   // [1]
                         float* __restrict__ out) {
    const int g = blockIdx.x;
    const int lane = threadIdx.x;  // blockDim == 32
    float s = H[g * H_DIM + lane] * W2[lane]
            + H[g * H_DIM + 32 + lane] * W2[32 + lane];
    #pragma unroll
    for (int off = 16; off > 0; off >>= 1)
        s += __shfl_down(s, off, 32);
    if (lane == 0) out[g] = s + b2[0];
}

extern "C" void kernel_launch(void* const* d_in, const int* in_sizes, int n_in,
                              void* d_out, int out_size, void* d_ws, size_t ws_size,
                              hipStream_t stream) {
    const float*     x     = (const float*)d_in[0];
    const long long* batch = (const long long*)d_in[1];
    const float*     W1    = (const float*)d_in[2];
    const float*     b1    = (const float*)d_in[3];
    const float*     W2    = (const float*)d_in[4];
    const float*     b2    = (const float*)d_in[5];
    float*           out   = (float*)d_out;

    const int n = in_sizes[1];     // number of nodes (batch length)
    const int G = out_size;        // graphs (256)

    float* sums   = (float*)d_ws;                    // [G,128]
    float* counts = sums + (size_t)G * D_DIM;        // [G]
    float* Hbuf   = counts + G;                      // [G,64]

    // Zero accumulators every launch (deterministic; harness poisons ws).
    const int zn = G * D_DIM + G;
    psr_zero<<<(zn + 255) / 256, 256, 0, stream>>>(sums, zn);

    // Bandwidth-bound segmented sum.
    const int RPW = 512;
    const int waves  = (n + RPW - 1) / RPW;
    const int blocks = (waves + 7) / 8;
    psr_segsum<<<blocks, 256, 0, stream>>>(x, batch, sums, counts, n, RPW);

    // WMMA MLP head: 64 tiles -> 64 waves -> 8 blocks of 8 waves.
    const int mTiles = G / 16;
    const int totWaves = mTiles * 4;
    psr_mlp1_wmma<<<(totWaves + 7) / 8, 256, 0, stream>>>(sums, counts, W1, b1,
                                                          Hbuf, mTiles);

    // Final matvec + bias.
    psr_mlp2<<<G, 32, 0, stream>>>(Hbuf, W2, b2, out);
}